// AdditiveAttention_90787018703176
// MI455X (gfx1250) — compile-verified
//
#include <hip/hip_runtime.h>
#include <math.h>

// Problem constants (match reference)
#define B_  16
#define Q_  64
#define K_  1024
#define D_  256
#define H_  256
#define NEGBIG (-1000000.0f)

#define QT        4            // q rows per scores block
#define HT        32           // h chunk (columns staged per async chunk)
#define SKT_STRIDE 257         // 257 % 64 == 1  -> conflict-free scatter & read
#define OPS_PER_CHUNK 32       // async b32 ops per wave per chunk

typedef __attribute__((ext_vector_type(2))) float v2f;
typedef __attribute__((ext_vector_type(8))) float v8f;

#define AS1 __attribute__((address_space(1)))
#define AS3 __attribute__((address_space(3)))

static __device__ __forceinline__ float fast_tanh(float x) {
#if __has_builtin(__builtin_amdgcn_tanhf)
    return __builtin_amdgcn_tanhf(x);
#else
    return tanhf(x);
#endif
}

// One b32 element: global -> LDS, asynchronous (per-lane addresses on both
// sides, so this is a transposing copy). Falls back to a synchronous copy.
static __device__ __forceinline__ void async_copy_b32(const float* g, float* l) {
#if __has_builtin(__builtin_amdgcn_global_load_async_to_lds_b32)
    __builtin_amdgcn_global_load_async_to_lds_b32((AS1 int*)g, (AS3 int*)l, 0, 0);
#else
    *l = *g;
#endif
}

template <int N>
static __device__ __forceinline__ void wait_async() {
#if __has_builtin(__builtin_amdgcn_s_wait_asynccnt)
    __builtin_amdgcn_s_wait_asynccnt(N);
#else
    asm volatile("s_wait_asynccnt %0" ::"n"(N) : "memory");
#endif
}

// ---------------------------------------------------------------------------
// Generic row-major f32 GEMM on the matrix pipe: C[M,N] = A[M,Kd] * Bm[Kd,N]
// One wave32 computes one 16x16 tile of C with V_WMMA_F32_16X16X4_F32,
// two independent accumulator chains to hide XDL latency.
// Batched via blockIdx.y + element strides (0 strides => single GEMM).
// ---------------------------------------------------------------------------
__global__ __launch_bounds__(256) void gemm_f32_wmma(
    const float* __restrict__ A, const float* __restrict__ Bm,
    float* __restrict__ C, int M, int N, int Kd,
    long batchStrideA, long batchStrideB, long batchStrideC)
{
    const long batch = blockIdx.y;
    A  += batch * batchStrideA;
    Bm += batch * batchStrideB;
    C  += batch * batchStrideC;

    const int gwave = (blockIdx.x * (int)blockDim.x + (int)threadIdx.x) >> 5;
    const int lane  = threadIdx.x & 31;
    const int hl    = lane >> 4;   // 0 => K k..k+1 / rows 0..7, 1 => K k+2..k+3 / rows 8..15
    const int l     = lane & 15;

    const int tiles_n = N >> 4;
    const int tiles_m = M >> 4;
    const int tile_m  = gwave / tiles_n;
    const int tile_n  = gwave % tiles_n;
    if (tile_m >= tiles_m) return;   // uniform per wave

    const float* arow = A  + (long)(tile_m * 16 + l) * Kd + 2 * hl;
    const float* bcol = Bm + (long)(2 * hl) * N + tile_n * 16 + l;

    v8f c0 = {};
    v8f c1 = {};
    for (int k = 0; k < Kd; k += 8) {
        v2f a0, b0, a1, b1;
        a0.x = arow[k];     a0.y = arow[k + 1];
        b0.x = bcol[(long)k * N];       b0.y = bcol[(long)(k + 1) * N];
        a1.x = arow[k + 4]; a1.y = arow[k + 5];
        b1.x = bcol[(long)(k + 4) * N]; b1.y = bcol[(long)(k + 5) * N];
        c0 = __builtin_amdgcn_wmma_f32_16x16x4_f32(false, a0, false, b0,
                                                   (short)0, c0, false, false);
        c1 = __builtin_amdgcn_wmma_f32_16x16x4_f32(false, a1, false, b1,
                                                   (short)0, c1, false, false);
    }
    v8f c = c0 + c1;

    float* crow = C + (long)(tile_m * 16 + 8 * hl) * N + tile_n * 16 + l;
#pragma unroll
    for (int r = 0; r < 8; ++r) crow[(long)r * N] = c[r];
}

// ---------------------------------------------------------------------------
// Fused additive-attention scores + masked softmax, async-LDS staged.
// Block = (batch b, group of QT=4 q rows), 256 threads (8 wave32).
// k_proj tiles are staged transposed into LDS (skT[h][k], stride 257) with
// GLOBAL_LOAD_ASYNC_TO_LDS_B32, double-buffered against the tanh compute.
// Each thread owns one k of a 256-wide k-tile and carries QT accumulators, so
// every staged k value is reused QT times and no cross-lane reduction needed.
// ---------------------------------------------------------------------------
__global__ __launch_bounds__(256) void scores_softmax(
    const float* __restrict__ qproj,   // [B,Q,H]
    const float* __restrict__ kproj,   // [B,K,H]
    const float* __restrict__ wv,      // [H]
    const int*   __restrict__ valid,   // [B]
    float*       __restrict__ attn)    // [B,Q,K]
{
    __shared__ float skT[2][HT * SKT_STRIDE];   // transposed k tile, double buffer
    __shared__ float s_q[QT * H_];
    __shared__ float s_wv[H_];
    __shared__ float s_sc[QT * K_];
    __shared__ float s_red[8];

    const int b   = blockIdx.x >> 4;          // B_ = 16 groups of 16 blocks
    const int qg  = blockIdx.x & 15;          // Q_/QT = 16
    const int q0  = qg * QT;
    const int tid = threadIdx.x;
    const int wave = tid >> 5;
    const int lane = tid & 31;

    s_wv[tid] = wv[tid];
#pragma unroll
    for (int q = 0; q < QT; ++q)
        s_q[q * H_ + tid] = qproj[((long)b * Q_ + q0 + q) * H_ + tid];

    const int vlen = valid[b];
    const float* kb = kproj + (long)b * K_ * H_;

    // chunk c = kt*8 + hc : k-tile kt in [0,4), h-chunk hc in [0,8)
    // stage chunk: skT[buf][h][kk] = kb[(kt*256+kk)*H_ + hc*32 + h]
    // wave w stages rows kk = w*32 .. w*32+31 (one async b32 op per row,
    // 32 contiguous global floats per op, conflict-free LDS scatter).
    auto issue_chunk = [&](int c, int buf) {
        const int kt = c >> 3, hc = c & 7;
        const float* g0 = kb + (long)(kt * 256 + wave * 32) * H_ + hc * HT + lane;
        float* l0 = &skT[buf][lane * SKT_STRIDE + wave * 32];
#pragma unroll
        for (int i = 0; i < 32; ++i)
            async_copy_b32(g0 + (long)i * H_, l0 + i);
    };

    float acc[QT];
    issue_chunk(0, 0);
    __syncthreads();   // s_q / s_wv ready (also covers first-iteration buffers)

    for (int c = 0; c < 32; ++c) {
        const int kt = c >> 3, hc = c & 7;
        const int buf = c & 1;
        if (c < 31) issue_chunk(c + 1, buf ^ 1);
        if (c < 31) wait_async<OPS_PER_CHUNK>();   // chunk c landed, c+1 in flight
        else        wait_async<0>();
        __syncthreads();

        if (hc == 0) {
#pragma unroll
            for (int q = 0; q < QT; ++q) acc[q] = 0.0f;
        }
#pragma unroll
        for (int h = 0; h < HT; ++h) {
            const float kv  = skT[buf][h * SKT_STRIDE + tid];
            const int   hh  = hc * HT + h;
            const float wvh = s_wv[hh];
#pragma unroll
            for (int q = 0; q < QT; ++q)
                acc[q] = fmaf(wvh, fast_tanh(s_q[q * H_ + hh] + kv), acc[q]);
        }
        if (hc == 7) {
            const int k = kt * 256 + tid;
            const bool ok = k < vlen;
#pragma unroll
            for (int q = 0; q < QT; ++q)
                s_sc[q * K_ + k] = ok ? acc[q] : NEGBIG;
        }
        __syncthreads();   // buffer free before it is re-issued
    }

    // ---- masked softmax per q row over K_ = 1024; each thread owns 4 ----
    for (int q = 0; q < QT; ++q) {
        const float* sc = &s_sc[q * K_];
        const float v0 = sc[tid];
        const float v1 = sc[tid + 256];
        const float v2 = sc[tid + 512];
        const float v3 = sc[tid + 768];

        float m = fmaxf(fmaxf(v0, v1), fmaxf(v2, v3));
#pragma unroll
        for (int off = 16; off > 0; off >>= 1)
            m = fmaxf(m, __shfl_down(m, off, 32));
        if (lane == 0) s_red[wave] = m;
        __syncthreads();
        if (tid == 0) {
            float mm = s_red[0];
#pragma unroll
            for (int w = 1; w < 8; ++w) mm = fmaxf(mm, s_red[w]);
            s_red[0] = mm;
        }
        __syncthreads();
        m = s_red[0];

        const float e0 = __expf(v0 - m);
        const float e1 = __expf(v1 - m);
        const float e2 = __expf(v2 - m);
        const float e3 = __expf(v3 - m);
        float s = (e0 + e1) + (e2 + e3);
#pragma unroll
        for (int off = 16; off > 0; off >>= 1)
            s += __shfl_down(s, off, 32);
        __syncthreads();               // everyone has read s_red[0]
        if (lane == 0) s_red[wave] = s;
        __syncthreads();
        if (tid == 0) {
            float ss = 0.0f;
#pragma unroll
            for (int w = 0; w < 8; ++w) ss += s_red[w];
            s_red[0] = ss;
        }
        __syncthreads();
        const float inv = 1.0f / s_red[0];

        float* arow = attn + ((long)b * Q_ + q0 + q) * K_;
        arow[tid]       = e0 * inv;
        arow[tid + 256] = e1 * inv;
        arow[tid + 512] = e2 * inv;
        arow[tid + 768] = e3 * inv;
        __syncthreads();               // s_red reused by next q row
    }
}

// ---------------------------------------------------------------------------
extern "C" void kernel_launch(void* const* d_in, const int* in_sizes, int n_in,
                              void* d_out, int out_size, void* d_ws, size_t ws_size,
                              hipStream_t stream)
{
    const float* queries = (const float*)d_in[0];   // [B,Q,D]
    const float* keys    = (const float*)d_in[1];   // [B,K,D]
    const float* values  = (const float*)d_in[2];   // [B,K,D]
    const int*   valid   = (const int*)  d_in[3];   // [B]
    const float* Wq      = (const float*)d_in[4];   // [D,H]
    const float* Wk      = (const float*)d_in[5];   // [D,H]
    const float* wv      = (const float*)d_in[6];   // [H]
    float*       out     = (float*)d_out;           // [B,Q,D]

    float* qproj = (float*)d_ws;                          //  1 MB  [B*Q, H]
    float* kproj = qproj + (size_t)B_ * Q_ * H_;          // 16 MB  [B*K, H]
    float* attn  = kproj + (size_t)B_ * K_ * H_;          //  4 MB  [B*Q, K]

    // 1) q_proj = queries @ W_q   (flat GEMM [B*Q, D] x [D, H])
    {
        const int M = B_ * Q_, N = H_, Kd = D_;
        const int tiles = (M / 16) * (N / 16);            // 1024 waves
        gemm_f32_wmma<<<dim3((tiles + 7) / 8, 1), 256, 0, stream>>>(
            queries, Wq, qproj, M, N, Kd, 0, 0, 0);
    }
    // 2) k_proj = keys @ W_k      (flat GEMM [B*K, D] x [D, H])
    {
        const int M = B_ * K_, N = H_, Kd = D_;
        const int tiles = (M / 16) * (N / 16);            // 16384 waves
        gemm_f32_wmma<<<dim3((tiles + 7) / 8, 1), 256, 0, stream>>>(
            keys, Wk, kproj, M, N, Kd, 0, 0, 0);
    }
    // 3) scores + masked softmax -> attn   (16 batches x 16 q-groups)
    scores_softmax<<<dim3(B_ * (Q_ / QT)), 256, 0, stream>>>(
        qproj, kproj, wv, valid, attn);

    // 4) out[b] = attn[b] @ values[b]   (batched GEMM, 64 tiles/batch)
    {
        const int M = Q_, N = D_, Kd = K_;
        const int tiles = (M / 16) * (N / 16);            // 64 waves per batch
        gemm_f32_wmma<<<dim3(tiles / 8, B_), 256, 0, stream>>>(
            attn, values, out, M, N, Kd,
            (long)Q_ * K_, (long)K_ * D_, (long)Q_ * D_);
    }
}